// StructureElementLayer_34883724378859
// MI455X (gfx1250) — compile-verified
//
#include <hip/hip_runtime.h>

// CDNA5 / gfx1250: wave32, fp32 WMMA path.
typedef float v2f __attribute__((ext_vector_type(2)));
typedef float v8f __attribute__((ext_vector_type(8)));

#define NPTS   4096
#define NELEM  64
#define NBATCH 64
#define WAVES  8                  // 256 threads / wave32
#define TILES_PER_WAVE (NPTS / 16 / WAVES)   // 32, compile-time uniform trip count

// out[b, e] = sum_p exp( -( s2x*(cx-x)^2*mask_x + s2y*(cy-y)^2*mask_y + zz ) )
// Rank-8 fp32 GEMM per 16x16 (element x point) tile:
//   A_e = -log2e * [s2x*cx^2, -2*s2x*cx, s2y*cy^2, -2*s2y*cy, s2x, s2y, 1, 0]
//   B_p =          [mx, x~, my, y~, x~^2, y~^2, zz, 0]
// so the WMMA accumulator directly holds -log2e*q and feeds v_exp_f32 (exp2)
// with no per-tile scaling VALU. Mask features reproduce nansum exactly.
__global__ __launch_bounds__(256) void structure_element_kernel(
    const float* __restrict__ X,   // (64, 4096, 3)
    const float* __restrict__ C,   // (64, 2) centres
    const float* __restrict__ S,   // (64, 2) sharpness
    float* __restrict__ out)       // (64, 64)
{
    const int b   = blockIdx.x;        // batch
    const int et  = blockIdx.y;        // element tile (16 elements each)
    const int tid = threadIdx.x;
    const int w   = tid >> 5;          // wave id 0..7
    const int l   = tid & 31;          // lane id
    const int m   = l & 15;            // M (A rows) / N (B cols) index
    const bool hi = (l >= 16);         // lanes 16..31 hold K=2,3 per ISA layout

    // ---- A-matrix features for element e = et*16 + m (loop-invariant) ----
    const int e   = et * 16 + m;
    const float cx  = C[2*e + 0], cy = C[2*e + 1];
    const float sx  = S[2*e + 0], sy = S[2*e + 1];
    const float NL2E = -1.4426950408889634f;           // -log2(e), folded into A
    const float s2x = NL2E * sx * sx, s2y = NL2E * sy * sy;

    // WMMA #0 covers K=0..3, WMMA #1 covers K=4..7.
    // A 16x4 f32 layout: lane m holds (K=kb, K=kb+1) in (v.x, v.y), kb = hi?2:0.
    v2f a0, a1;
    a0.x = hi ? (s2y * cy * cy)   : (s2x * cx * cx);
    a0.y = hi ? (-2.f * s2y * cy) : (-2.f * s2x * cx);
    a1.x = hi ? NL2E : s2x;
    a1.y = hi ? 0.f  : s2y;

    // Running per-lane sum of exp(-q) over this wave's point tiles.
    // C/D layout: VGPR r, lanes 0-15 -> row M=r, lanes 16-31 -> M=r+8; col N = l&15.
    v8f acc = {};

    const float* xb = X + (size_t)b * NPTS * 3;

    // 256 point-tiles of 16; wave w takes tiles w, w+8, ...
    // Compile-time trip count -> no EXEC masking, WMMA sees all-ones EXEC.
    #pragma unroll 2
    for (int it = 0; it < TILES_PER_WAVE; ++it) {
        const int t = it * WAVES + w;
        const int p = t * 16 + m;               // this lane's point (col N of B)
        const float xv = xb[3*p + 0];
        const float yv = xb[3*p + 1];
        const float zv = xb[3*p + 2];

        // nansum semantics via mask features
        const float mx = (xv == xv) ? 1.f : 0.f;
        const float xt = (xv == xv) ? xv  : 0.f;
        const float my = (yv == yv) ? 1.f : 0.f;
        const float yt = (yv == yv) ? yv  : 0.f;
        const float zz = (zv == zv) ? zv * zv : 0.f;

        // B 4x16 f32 layout mirrors A: lane holds rows (kb, kb+1) for its column.
        v2f b0, b1;
        b0.x = hi ? my  : mx;
        b0.y = hi ? yt  : xt;
        b1.x = hi ? zz  : xt * xt;
        b1.y = hi ? 0.f : yt * yt;

        // q' = -log2e * q  via two K=4 fp32 WMMA steps (exact fp32 accumulate)
        v8f q = {};
        q = __builtin_amdgcn_wmma_f32_16x16x4_f32(false, a0, false, b0,
                                                  (short)0, q, false, false);
        q = __builtin_amdgcn_wmma_f32_16x16x4_f32(false, a1, false, b1,
                                                  (short)0, q, false, false);

        #pragma unroll
        for (int i = 0; i < 8; ++i)
            acc[i] += __builtin_amdgcn_exp2f(q[i]);   // exp(-q) = exp2(q')
    }

    // ---- reduce Σ_p across the 16 columns (xor masks 1,2,4,8 never cross bit 4,
    //      so full-width shuffles stay inside each 16-lane half) ----
    #pragma unroll
    for (int i = 0; i < 8; ++i) {
        float v = acc[i];
        v += __shfl_xor(v, 1);
        v += __shfl_xor(v, 2);
        v += __shfl_xor(v, 4);
        v += __shfl_xor(v, 8);
        acc[i] = v;
    }

    // ---- combine waves via LDS (deterministic, fixed order) ----
    __shared__ float red[WAVES][16];
    if (m == 0) {                       // lane 0 -> rows 0..7, lane 16 -> rows 8..15
        const int rb = hi ? 8 : 0;
        #pragma unroll
        for (int i = 0; i < 8; ++i) red[w][rb + i] = acc[i];
    }
    __syncthreads();

    if (tid < 16) {
        float s = 0.f;
        #pragma unroll
        for (int ww = 0; ww < WAVES; ++ww) s += red[ww][tid];
        out[b * NELEM + et * 16 + tid] = s;
    }
}

extern "C" void kernel_launch(void* const* d_in, const int* in_sizes, int n_in,
                              void* d_out, int out_size, void* d_ws, size_t ws_size,
                              hipStream_t stream) {
    (void)in_sizes; (void)n_in; (void)d_ws; (void)ws_size; (void)out_size;
    const float* X = (const float*)d_in[0];   // (64, 4096, 3) f32
    const float* C = (const float*)d_in[1];   // (64, 2) f32
    const float* S = (const float*)d_in[2];   // (64, 2) f32
    float* out = (float*)d_out;               // (64, 64) f32

    dim3 grid(NBATCH, NELEM / 16);            // 64 batches x 4 element tiles
    structure_element_kernel<<<grid, 256, 0, stream>>>(X, C, S, out);
}